// SpatialTransformer_7499012899136
// MI455X (gfx1250) — compile-verified
//
#include <hip/hip_runtime.h>

// SpatialTransformer: 3D trilinear grid-sample, border padding, align_corners,
// faithful to the reference's "only x-channel unnormalized" quirk.
//
// Memory-bound streaming gather (~275 MB unique HBM traffic -> ~12us floor at
// 23.3 TB/s). No matmul structure => WMMA inapplicable (per-lane unique
// weights/addresses). gfx1250 features that pay: NT temporal hints on the
// single-use flow/out streams (keep img resident in the 192MB L2) and
// global_prefetch_b8 on the statistically predicted gather rows.
//
// Index math is folded entirely into the launch grid: block(56,4)=224 threads
// = exactly 7 wave32; grid(H/4, D, B). Zero integer divisions in the kernel.

namespace {

constexpr int D   = 160;
constexpr int H   = 192;
constexpr int W   = 224;
constexpr int HW  = H * W;        // 43008
constexpr int DHW = D * HW;       // 6881280

typedef __attribute__((ext_vector_type(4))) float v4f;

// Unaligned-capable pair load (float at p, p+1); 4-byte aligned source.
struct __attribute__((packed, aligned(4))) f2u { float lo, hi; };

__device__ __forceinline__ float lerpf(float a, float b, float t) {
  return __builtin_fmaf(t, b - a, a);  // a + t*(b-a)
}

__global__ __launch_bounds__(224)
void st3d_trilinear(const float* __restrict__ img,
                    const float* __restrict__ flow,
                    float* __restrict__ out) {
  // block(56,4): threadIdx.x = x-group (4 voxels), threadIdx.y = y sub-row.
  // grid(H/4, D, B).
  const int x = threadIdx.x * 4;
  const int y = blockIdx.x * 4 + threadIdx.y;
  const int z = blockIdx.y;
  const int b = blockIdx.z;

  const int s = z * HW + y * W + x;

  const float* imgB  = img  + (size_t)b * DHW;
  const float* flowB = flow + (size_t)b * (3 * (size_t)DHW) + s;

  // Predicted gather neighborhood: px ~= clamp((x+1+fx)*111.5, 0, 223) with
  // fx ~ N(0,1); py ~= y, pz ~= z. Prefetch the two y-rows at the predicted x
  // so the (L2-resident) gathers warm L0 while the flow b128 loads are in
  // flight. gfx1250: lowers to global_prefetch_b8.
  {
    float predx = (float)(x + 2) * (0.5f * (float)(W - 1));
    predx = fminf(predx, (float)(W - 1));
    int pxi = (int)predx;
    const float* prow = imgB + (size_t)z * HW + (size_t)y * W + pxi;
    __builtin_prefetch(prow, 0, 3);
    __builtin_prefetch(prow + W, 0, 3);
  }

  // Single-use streams: non-temporal so the 192MB L2 keeps img resident.
  v4f f0 = __builtin_nontemporal_load((const v4f*)(flowB));            // flow ch0 (x)
  v4f f1 = __builtin_nontemporal_load((const v4f*)(flowB + DHW));      // flow ch1 (y)
  v4f f2 = __builtin_nontemporal_load((const v4f*)(flowB + 2 * DHW));  // flow ch2 (z)

  float fx4[4] = {f0.x, f0.y, f0.z, f0.w};
  float fy4[4] = {f1.x, f1.y, f1.z, f1.w};
  float fz4[4] = {f2.x, f2.y, f2.z, f2.w};
  float r[4];

#pragma unroll
  for (int j = 0; j < 4; ++j) {
    // Reference coordinate math (x-channel only goes through grid unnorm).
    float px = (((float)(x + j) + fx4[j]) + 1.0f) * (0.5f * (float)(W - 1));
    float py = (float)y + fy4[j];
    float pz = (float)z + fz4[j];
    // Border clamp, align_corners=True.
    px = fminf(fmaxf(px, 0.0f), (float)(W - 1));
    py = fminf(fmaxf(py, 0.0f), (float)(H - 1));
    pz = fminf(fmaxf(pz, 0.0f), (float)(D - 1));

    // Shifted-base trick: base = min(floor(p), dim-2), frac = p - base.
    // Identical to (floor, min(floor+1,dim-1), frac) including the exact
    // p == dim-1 border case (frac becomes 1.0 and selects the last sample),
    // and lets each x-pair come from one 8-byte load.
    int xb = (int)px; xb = xb > (W - 2) ? (W - 2) : xb;   // px >= 0 -> trunc==floor
    int yb = (int)py; yb = yb > (H - 2) ? (H - 2) : yb;
    int zb = (int)pz; zb = zb > (D - 2) ? (D - 2) : zb;
    float tx = px - (float)xb;
    float ty = py - (float)yb;
    float tz = pz - (float)zb;

    const float* p = imgB + (size_t)zb * HW + (size_t)yb * W + xb;
    f2u c00 = *(const f2u*)(p);            // (z0,y0,x0..x0+1)
    f2u c01 = *(const f2u*)(p + W);        // (z0,y1,*)
    f2u c10 = *(const f2u*)(p + HW);       // (z1,y0,*)
    f2u c11 = *(const f2u*)(p + HW + W);   // (z1,y1,*)

    float a0 = lerpf(c00.lo, c00.hi, tx);
    float a1 = lerpf(c01.lo, c01.hi, tx);
    float a2 = lerpf(c10.lo, c10.hi, tx);
    float a3 = lerpf(c11.lo, c11.hi, tx);
    float e0 = lerpf(a0, a1, ty);
    float e1 = lerpf(a2, a3, ty);
    r[j] = lerpf(e0, e1, tz);
  }

  v4f res = {r[0], r[1], r[2], r[3]};
  __builtin_nontemporal_store(res, (v4f*)(out + (size_t)b * DHW + s));
}

}  // namespace

extern "C" void kernel_launch(void* const* d_in, const int* in_sizes, int n_in,
                              void* d_out, int out_size, void* d_ws, size_t ws_size,
                              hipStream_t stream) {
  (void)in_sizes; (void)n_in; (void)out_size; (void)d_ws; (void)ws_size;
  const float* img  = (const float*)d_in[0];
  const float* flow = (const float*)d_in[1];
  float* out        = (float*)d_out;

  dim3 block(W / 4, 4);        // 56 x 4 = 224 threads = 7 wave32, no partial wave
  dim3 grid(H / 4, D, 2);      // y-tiles, z, batch
  hipLaunchKernelGGL(st3d_trilinear, grid, block, 0, stream, img, flow, out);
}